// GNNModel_63617055588405
// MI455X (gfx1250) — compile-verified
//
#include <hip/hip_runtime.h>

typedef __attribute__((ext_vector_type(2))) float v2f;
typedef __attribute__((ext_vector_type(8))) float v8f;

#define DH 64
#define DOUTC 16

// ---------------------------------------------------------------------------
// Degree / normalization
// ---------------------------------------------------------------------------
__global__ void init_deg_kernel(float* __restrict__ deg, int n) {
    int i = blockIdx.x * blockDim.x + threadIdx.x;
    if (i < n) deg[i] = 1.0f;                    // self-loop contributes 1
}

__global__ void accum_deg_kernel(const int* __restrict__ dst,
                                 float* __restrict__ deg, int e) {
    int i = blockIdx.x * blockDim.x + threadIdx.x;
    if (i < e) unsafeAtomicAdd(&deg[dst[i]], 1.0f);
}

__global__ void rsqrt_kernel(float* __restrict__ deg, int n) {
    int i = blockIdx.x * blockDim.x + threadIdx.x;
    if (i < n) deg[i] = rsqrtf(deg[i]);          // in-place deg -> deg^{-1/2}
}

// ---------------------------------------------------------------------------
// Dense transform + fused aggregation-init.
//   hW[i,:]  = (RELU ? relu(Hin[i,:]) : Hin[i,:]) @ W^T
//   agg[i,:] = dinv[i]^2 * hW[i,:] + bias        (self-loop term + bias)
// One wave per 16x16 output tile; 4 waves (128 thr) per block cover 16 rows
// x 64 cols. K advances 4 per V_WMMA_F32_16X16X4_F32 (16 WMMAs / tile).
// A: lanes 0-15 -> M=lane, K=(k,k+1); lanes 16-31 -> M=lane-16, K=(k+2,k+3)
// B: lanes 0-15 -> N=lane, K=(k,k+1); lanes 16-31 -> N=lane-16, K=(k+2,k+3)
// B[k][n] = W[n][k] (W row-major [out,in]) -> contiguous float2 load.
// RELU is compile-time: no-relu layer pays zero extra VALU, relu layers pay
// one v_max_num per element (no cndmask chains).
// ---------------------------------------------------------------------------
template <int RELU>
__global__ __launch_bounds__(128)
void gcn_gemm_wmma(const float* __restrict__ Hin, const float* __restrict__ W,
                   const float* __restrict__ dinv, const float* __restrict__ bias,
                   float* __restrict__ hW, float* __restrict__ agg) {
    const int lane    = threadIdx.x & 31;
    const int wave    = threadIdx.x >> 5;        // col tile 0..3
    const int rowbase = blockIdx.x * 16;
    const int colbase = wave * 16;
    const int half    = lane >> 4;               // 0 or 1
    const int l15     = lane & 15;

    const float* arow = Hin + (size_t)(rowbase + l15) * DH;
    const float* brow = W   + (size_t)(colbase + l15) * DH;

    v8f acc = {};
#pragma unroll
    for (int k = 0; k < DH; k += 4) {
        float2 av = *(const float2*)(arow + k + 2 * half);
        float2 bv = *(const float2*)(brow + k + 2 * half);
        if (RELU) { av.x = fmaxf(av.x, 0.0f); av.y = fmaxf(av.y, 0.0f); }
        v2f a; a.x = av.x; a.y = av.y;
        v2f b; b.x = bv.x; b.y = bv.y;
        acc = __builtin_amdgcn_wmma_f32_16x16x4_f32(
            /*neg_a=*/false, a, /*neg_b=*/false, b,
            /*c_mod=*/(short)0, acc, /*reuse_a=*/false, /*reuse_b=*/false);
    }

    // D: VGPR v -> row rowbase+v (lanes 0-15) / rowbase+v+8 (lanes 16-31), col=l15
    const int   col = colbase + l15;
    const float bc  = bias[col];
#pragma unroll
    for (int v = 0; v < 8; ++v) {
        const int    row = rowbase + v + half * 8;
        const size_t idx = (size_t)row * DH + col;
        const float  di  = dinv[row];
        hW[idx]  = acc[v];
        agg[idx] = di * di * acc[v] + bc;
    }
}

// ---------------------------------------------------------------------------
// Edge scatter: agg[dst,:] += dinv[src]*dinv[dst] * hW[src,:]
// 16 lanes per edge, 4 channels each (float4 gather + 4 hw f32 atomics).
// Working set is L2-resident (192 MB) -> bound by L2 atomic throughput.
// ---------------------------------------------------------------------------
__global__ void agg_edges_kernel(const int* __restrict__ src,
                                 const int* __restrict__ dst,
                                 const float* __restrict__ dinv,
                                 const float* __restrict__ hW,
                                 float* __restrict__ agg, int e) {
    long long t = (long long)blockIdx.x * blockDim.x + threadIdx.x;
    int eid = (int)(t >> 4);
    if (eid >= e) return;
    int c = ((int)t & 15) * 4;
    int s = src[eid], d = dst[eid];
    float w = dinv[s] * dinv[d];
    float4 v = *(const float4*)(hW + (size_t)s * DH + c);
    float* o = agg + (size_t)d * DH + c;
    unsafeAtomicAdd(o + 0, w * v.x);
    unsafeAtomicAdd(o + 1, w * v.y);
    unsafeAtomicAdd(o + 2, w * v.z);
    unsafeAtomicAdd(o + 3, w * v.w);
}

// ---------------------------------------------------------------------------
// Global mean pool (ReLU fused on input) + counts
// ---------------------------------------------------------------------------
__global__ void pool_zero_kernel(float* __restrict__ sums, float* __restrict__ cnt, int b) {
    int i = blockIdx.x * blockDim.x + threadIdx.x;
    if (i < b * DH) sums[i] = 0.0f;
    if (i < b) cnt[i] = 0.0f;
}

__global__ void pool_accum_kernel(const float* __restrict__ h,
                                  const int* __restrict__ batch,
                                  float* __restrict__ sums,
                                  float* __restrict__ cnt, int n) {
    int idx = blockIdx.x * blockDim.x + threadIdx.x;
    if (idx >= n * DH) return;
    int i = idx >> 6, c = idx & 63;
    int g = batch[i];
    unsafeAtomicAdd(&sums[g * DH + c], fmaxf(h[idx], 0.0f));
    if (c == 0) unsafeAtomicAdd(&cnt[g], 1.0f);
}

// ---------------------------------------------------------------------------
// out[b,o] = (sums[b,:]/max(cnt,1)) . Wl[o,:] + bl[o]   (tiny, VALU is fine)
// ---------------------------------------------------------------------------
__global__ void final_linear_kernel(const float* __restrict__ sums,
                                    const float* __restrict__ cnt,
                                    const float* __restrict__ Wl,
                                    const float* __restrict__ bl,
                                    float* __restrict__ out, int b) {
    int idx = blockIdx.x * blockDim.x + threadIdx.x;
    if (idx >= b * DOUTC) return;
    int g = idx >> 4, o = idx & 15;
    float inv = 1.0f / fmaxf(cnt[g], 1.0f);
    float acc = bl[o];
    const float* p = sums + (size_t)g * DH;
    const float* w = Wl + (size_t)o * DH;
#pragma unroll 8
    for (int c = 0; c < DH; ++c) acc = fmaf(p[c] * inv, w[c], acc);
    out[idx] = acc;
}

// ---------------------------------------------------------------------------
extern "C" void kernel_launch(void* const* d_in, const int* in_sizes, int n_in,
                              void* d_out, int out_size, void* d_ws, size_t ws_size,
                              hipStream_t stream) {
    const float* x    = (const float*)d_in[0];
    const int*   edge = (const int*)d_in[1];
    const int*   batch= (const int*)d_in[2];
    const float* W1   = (const float*)d_in[3];
    const float* b1   = (const float*)d_in[4];
    const float* W2   = (const float*)d_in[5];
    const float* b2   = (const float*)d_in[6];
    const float* W3   = (const float*)d_in[7];
    const float* b3   = (const float*)d_in[8];
    const float* Wl   = (const float*)d_in[9];
    const float* bl   = (const float*)d_in[10];
    float* out = (float*)d_out;

    const int n = in_sizes[0] / DH;      // 50000
    const int e = in_sizes[1] / 2;       // 1250000
    const int b = out_size / DOUTC;      // 512
    const int* src = edge;               // edge_index[0]
    const int* dst = edge + e;           // edge_index[1]

    // workspace (f32): dinv[n] | hWbuf[n*64] | agg0[n*64] | agg1[n*64] | sums[b*64] | cnt[b]
    // (~39 MB). GEMM l reads aggP, writes hWbuf+aggQ (P!=Q); edge pass reads
    // hWbuf, atomically updates aggQ. No kernel reads and writes the same buf.
    float* ws    = (float*)d_ws;
    float* dinv  = ws;
    float* hWbuf = dinv + n;
    float* agg0  = hWbuf + (size_t)n * DH;
    float* agg1  = agg0 + (size_t)n * DH;
    float* sums  = agg1 + (size_t)n * DH;
    float* cnt   = sums + (size_t)b * DH;

    const int T = 256;

    // deg -> dinv
    init_deg_kernel<<<(n + T - 1) / T, T, 0, stream>>>(dinv, n);
    accum_deg_kernel<<<(e + T - 1) / T, T, 0, stream>>>(dst, dinv, e);
    rsqrt_kernel<<<(n + T - 1) / T, T, 0, stream>>>(dinv, n);

    const int nc_grid = (n * DH + T - 1) / T;
    const int eg_grid = (int)(((long long)e * 16 + T - 1) / T);

    // Layer 1: x -> agg0
    gcn_gemm_wmma<0><<<n / 16, 128, 0, stream>>>(x, W1, dinv, b1, hWbuf, agg0);
    agg_edges_kernel<<<eg_grid, T, 0, stream>>>(src, dst, dinv, hWbuf, agg0, e);
    // Layer 2: agg0 -> agg1 (ReLU fused into GEMM load)
    gcn_gemm_wmma<1><<<n / 16, 128, 0, stream>>>(agg0, W2, dinv, b2, hWbuf, agg1);
    agg_edges_kernel<<<eg_grid, T, 0, stream>>>(src, dst, dinv, hWbuf, agg1, e);
    // Layer 3: agg1 -> agg0
    gcn_gemm_wmma<1><<<n / 16, 128, 0, stream>>>(agg1, W3, dinv, b3, hWbuf, agg0);
    agg_edges_kernel<<<eg_grid, T, 0, stream>>>(src, dst, dinv, hWbuf, agg0, e);

    // mean pool (ReLU fused) + classifier
    pool_zero_kernel<<<(b * DH + T - 1) / T, T, 0, stream>>>(sums, cnt, b);
    pool_accum_kernel<<<nc_grid, T, 0, stream>>>(agg0, batch, sums, cnt, n);
    final_linear_kernel<<<(b * DOUTC + T - 1) / T, T, 0, stream>>>(sums, cnt, Wl, bl, out, b);
}